// StatefulLIF_88940182765806
// MI455X (gfx1250) — compile-verified
//
#include <hip/hip_runtime.h>

// StatefulLIF forward_sequence for MI455X (gfx1250).
// x:[B,T,N] f32 -> spikes:[B,T,N], m_f:[B,N], tr_f:[B,N] (concat in d_out).
//
// Roofline: ~5 FLOP per element vs 8 B moved -> AI ~0.6 FLOP/B, pure
// bandwidth problem (128 MiB @ 23.3 TB/s ~ 5.8 us floor; x fits in 192 MB L2
// across graph replays). The recurrence is nonlinear (Heaviside + reset), so
// there is no matmul/scan reformulation -> WMMA inapplicable. The CDNA5
// leverage is the async global->LDS DMA path: 16 in-flight cachelines per
// wave tracked by ASYNCcnt, consumed via ds_load with split-counter waits,
// with zero VGPR cost for prefetch depth.

#define B_      8
#define T_      1024
#define N_      2048
#define BLOCK   64            // 2 waves; grid = B*N/BLOCK = 256 workgroups
#define STAGES  16            // circular LDS staging depth (ASYNCcnt <= 63 ok)

// GVS form: copies a dword from the SGPR base plus VGPR byte offset into LDS at lds_off; bumps ASYNCcnt.
// Keeps the 64-bit base uniform in SGPRs; per-lane address math is one
// 32-bit add per timestep.
__device__ __forceinline__ void async_load_f32(uint32_t lds_off, uint32_t voff,
                                               const float* sbase) {
    asm volatile("global_load_async_to_lds_b32 %0, %1, %2"
                 :
                 : "v"(lds_off), "v"(voff), "s"(sbase)
                 : "memory");
}

// LDS read of the staged value; s_wait_dscnt 0 both delivers the data and
// closes the WAR hazard before this slot is re-targeted by a new async copy.
__device__ __forceinline__ float lds_read_f32(uint32_t lds_off) {
    float v;
    asm volatile("ds_load_b32 %0, %1\n\t"
                 "s_wait_dscnt 0x0"
                 : "=v"(v)
                 : "v"(lds_off)
                 : "memory");
    return v;
}

__global__ __launch_bounds__(BLOCK) void lif_seq_kernel(
    const float* __restrict__ x,
    const float* __restrict__ m_in,
    const float* __restrict__ tr_in,
    float* __restrict__ spikes,
    float* __restrict__ m_out,
    float* __restrict__ tr_out)
{
    const int tid = threadIdx.x;
    const int g   = blockIdx.x * BLOCK + tid;   // 0 .. B*N-1
    const int b   = g >> 11;                    // / N_  (uniform per block)
    const int n0  = (blockIdx.x * BLOCK) & (N_ - 1);   // block-uniform

    // Wave-uniform 64-bit base for this block's (b, n0) column group.
    const float* xbase = x + (size_t)b * T_ * N_ + n0;
    float*       sp    = spikes + (size_t)b * T_ * N_ + n0 + tid;

    const float a_m   = 0.90483741803595952f;   // exp(-1/10)
    const float a_s   = 0.81873075307798182f;   // exp(-1/5)
    const float c_in  = 1.0f - a_m;
    const float v_th  = 1.0f;
    const float r_amt = 1.0f;                   // V_TH - V_RESET

    float m  = m_in[g];
    float tr = tr_in[g];

    // Dynamic LDS starts at offset 0 (no static LDS in this kernel).
    // Slot for (stage s, lane tid) at byte offset (s*BLOCK + tid)*4.
    const uint32_t lane_lds = (uint32_t)tid * 4u;
    const uint32_t lane_mem = (uint32_t)tid * 4u;        // byte off within row
    const uint32_t row_bytes = (uint32_t)N_ * 4u;        // 8192 B per timestep

    // Prologue: fill all STAGES slots.
#pragma unroll
    for (int s = 0; s < STAGES; ++s)
        async_load_f32(lane_lds + (uint32_t)s * (BLOCK * 4u),
                       lane_mem + (uint32_t)s * row_bytes, xbase);

    uint32_t voff = lane_mem + (uint32_t)STAGES * row_bytes;  // next to fetch

    int t = 0;
    // Main pipelined loop: steady state keeps STAGES async copies in flight.
    for (; t < T_ - STAGES; ++t) {
        // Oldest of STAGES outstanding copies (in-order completion) is done.
        asm volatile("s_wait_asynccnt 0xf" ::: "memory");
        const uint32_t off = lane_lds + (uint32_t)(t & (STAGES - 1)) * (BLOCK * 4u);
        const float xv = lds_read_f32(off);        // includes s_wait_dscnt 0
        async_load_f32(off, voff, xbase);          // refill slot for t+STAGES
        voff += row_bytes;

        m = fmaf(a_m, m, c_in * xv);
        const float s = (m > v_th) ? 1.0f : 0.0f;
        m  = fmaf(-r_amt, s, m);
        tr = fmaf(a_s, tr, s);
        __builtin_nontemporal_store(s, sp + (size_t)t * N_);  // write-once stream
    }

    // Drain: everything already issued; wait once, then consume the tail.
    asm volatile("s_wait_asynccnt 0x0" ::: "memory");
    for (; t < T_; ++t) {
        const uint32_t off = lane_lds + (uint32_t)(t & (STAGES - 1)) * (BLOCK * 4u);
        const float xv = lds_read_f32(off);

        m = fmaf(a_m, m, c_in * xv);
        const float s = (m > v_th) ? 1.0f : 0.0f;
        m  = fmaf(-r_amt, s, m);
        tr = fmaf(a_s, tr, s);
        __builtin_nontemporal_store(s, sp + (size_t)t * N_);
    }

    m_out[g]  = m;
    tr_out[g] = tr;
}

extern "C" void kernel_launch(void* const* d_in, const int* in_sizes, int n_in,
                              void* d_out, int out_size, void* d_ws, size_t ws_size,
                              hipStream_t stream) {
    (void)in_sizes; (void)n_in; (void)out_size; (void)d_ws; (void)ws_size;

    const float* x   = (const float*)d_in[0];   // [B,T,N]
    const float* m0  = (const float*)d_in[1];   // [B,N]
    const float* tr0 = (const float*)d_in[2];   // [B,N]

    float* spikes = (float*)d_out;                       // [B,T,N]
    float* m_out  = spikes + (size_t)B_ * T_ * N_;       // [B,N]
    float* tr_out = m_out  + (size_t)B_ * N_;            // [B,N]

    const int nthreads = B_ * N_;                        // 16384 lanes
    dim3 grid(nthreads / BLOCK);                         // 256 workgroups
    dim3 block(BLOCK);
    size_t lds_bytes = (size_t)STAGES * BLOCK * sizeof(float);  // 4 KB

    lif_seq_kernel<<<grid, block, lds_bytes, stream>>>(x, m0, tr0,
                                                       spikes, m_out, tr_out);
}